// Tmp_attention4_70841190580535
// MI455X (gfx1250) — compile-verified
//
#include <hip/hip_runtime.h>
#include <hip/hip_bf16.h>

typedef float v2f __attribute__((ext_vector_type(2)));
typedef float v8f __attribute__((ext_vector_type(8)));

#define B_ 4
#define N_ 384
#define CS_ 384
#define H_ 12
#define C_ 16
#define PQ_ 4
#define PV_ 8
#define BN_ (B_ * N_)  // 1536

// D = A(16x4) x B(4x16) + C, fp32 WMMA
__device__ inline v8f wmma4(v2f a, v2f b, v8f c) {
  return __builtin_amdgcn_wmma_f32_16x16x4_f32(false, a, false, b, (short)0, c,
                                               false, false);
}

// CDNA5 async global -> LDS copy (16B), tracked by ASYNCcnt.
// LDS address = low 32 bits of the generic pointer ({shared_base, offset32}).
__device__ inline void async_copy_b128(float* lds_dst, const float* gsrc) {
  unsigned ldsoff = (unsigned)(unsigned long long)lds_dst;
  asm volatile("global_load_async_to_lds_b128 %0, %1, off"
               :
               : "v"(ldsoff), "v"(gsrc)
               : "memory");
}
__device__ inline void wait_asynccnt0() {
  asm volatile("s_wait_asynccnt 0" ::: "memory");
}

// ---------------------------------------------------------------------------
// Generic GEMM: Cout[M x Nout] = A[M x K] @ W[K x Nout] + bias
// One wave per 16x16 output tile. K % 4 == 0, Nout % 16 == 0, M % 16 == 0.
// ---------------------------------------------------------------------------
__global__ void __launch_bounds__(128) gemm16_bias(
    const float* __restrict__ A, const float* __restrict__ W,
    const float* __restrict__ bias, float* __restrict__ Cout, int K, int Nout) {
  const int lane = threadIdx.x & 31;
  const int wave = threadIdx.x >> 5;
  const int m0 = blockIdx.x * 16;
  const int n0 = (blockIdx.y * 4 + wave) * 16;
  if (n0 >= Nout) return;  // wave-uniform: EXEC stays all-ones for live waves
  const int lh = lane & 15;
  const int hi = lane >> 4;

  const float* arow = A + (m0 + lh) * K + 2 * hi;
  const float* wcol = W + (2 * hi) * Nout + n0 + lh;

  v8f acc = {};
  for (int kk = 0; kk < K; kk += 4) {
    v2f a = *(const v2f*)(arow + kk);  // A[m0+lh][kk+2hi .. +1]
    v2f b;
    b.x = wcol[kk * Nout];           // W[kk+2hi][n0+lh]
    b.y = wcol[kk * Nout + Nout];    // W[kk+2hi+1][n0+lh]
    acc = wmma4(a, b, acc);
  }
  const float bv = bias[n0 + lh];
  float* crow = Cout + (m0 + 8 * hi) * Nout + n0 + lh;
#pragma unroll
  for (int r = 0; r < 8; r++) crow[r * Nout] = acc[r] + bv;
}

// ---------------------------------------------------------------------------
// Prep: rigid-apply, build augmented q'/k' (32-wide) and V' (48-wide), colbias
// q'[0:16]=2*scale*q, q'[16:28]=hw*q_pts ; k'[0:16]=k, k'[16:28]=k_pts
// V'[0:16]=v, V'[16:40]=v_pts ; colbias = -0.5*hw*|k_pts|^2 + INF*(mask-1)
// ---------------------------------------------------------------------------
__global__ void prep_kernel(
    const float* __restrict__ q, const float* __restrict__ kv,
    const float* __restrict__ qp_raw, const float* __restrict__ kvp_raw,
    const float* __restrict__ rot, const float* __restrict__ trans,
    const float* __restrict__ mask, const float* __restrict__ head_weights,
    float* __restrict__ qa, float* __restrict__ ka, float* __restrict__ va,
    float* __restrict__ colbias) {
  int idx = blockIdx.x * blockDim.x + threadIdx.x;  // b*N + n
  if (idx >= BN_) return;
  const int b = idx / N_, n = idx % N_;
  float R[9], T[3];
#pragma unroll
  for (int i = 0; i < 9; i++) R[i] = rot[idx * 9 + i];
#pragma unroll
  for (int i = 0; i < 3; i++) T[i] = trans[idx * 3 + i];

  const float qscale = 2.0f * 0.14433756729740643f;  // 2*sqrt(1/(3*C))
  const float hwscale = 0.13608276348795434f;        // sqrt(1/(3*PQ*9/2))
  const float mbias = 100000.0f * (mask[idx] - 1.0f);

  for (int h = 0; h < H_; h++) {
    const float hw = log1pf(expf(head_weights[h])) * hwscale;  // softplus
    float* qrow = qa + ((b * H_ + h) * N_ + n) * 32;
    float* krow = ka + ((b * H_ + h) * N_ + n) * 32;
    float* vrow = va + ((b * H_ + h) * N_ + n) * 48;
#pragma unroll
    for (int c = 0; c < 16; c++) {
      qrow[c] = qscale * q[idx * 192 + h * 16 + c];
      krow[c] = kv[idx * 384 + h * 32 + c];
      vrow[c] = kv[idx * 384 + h * 32 + 16 + c];
    }
#pragma unroll
    for (int p = 0; p < PQ_; p++) {  // q points: raw layout [3][H*PQ]
      float x = qp_raw[idx * 144 + 0 * 48 + h * PQ_ + p];
      float y = qp_raw[idx * 144 + 1 * 48 + h * PQ_ + p];
      float z = qp_raw[idx * 144 + 2 * 48 + h * PQ_ + p];
      float gx = R[0] * x + R[1] * y + R[2] * z + T[0];
      float gy = R[3] * x + R[4] * y + R[5] * z + T[1];
      float gz = R[6] * x + R[7] * y + R[8] * z + T[2];
      qrow[16 + p * 3 + 0] = hw * gx;
      qrow[16 + p * 3 + 1] = hw * gy;
      qrow[16 + p * 3 + 2] = hw * gz;
    }
    qrow[28] = 0.f; qrow[29] = 0.f; qrow[30] = 0.f; qrow[31] = 0.f;
    krow[28] = 0.f; krow[29] = 0.f; krow[30] = 0.f; krow[31] = 0.f;

    float ksq = 0.f;
#pragma unroll
    for (int p = 0; p < PQ_ + PV_; p++) {  // kv points: raw layout [3][H*12]
      float x = kvp_raw[idx * 432 + 0 * 144 + h * 12 + p];
      float y = kvp_raw[idx * 432 + 1 * 144 + h * 12 + p];
      float z = kvp_raw[idx * 432 + 2 * 144 + h * 12 + p];
      float gx = R[0] * x + R[1] * y + R[2] * z + T[0];
      float gy = R[3] * x + R[4] * y + R[5] * z + T[1];
      float gz = R[6] * x + R[7] * y + R[8] * z + T[2];
      if (p < PQ_) {
        krow[16 + p * 3 + 0] = gx;
        krow[16 + p * 3 + 1] = gy;
        krow[16 + p * 3 + 2] = gz;
        ksq += gx * gx + gy * gy + gz * gz;
      } else {
        const int pp = p - PQ_;
        vrow[16 + pp * 3 + 0] = gx;
        vrow[16 + pp * 3 + 1] = gy;
        vrow[16 + pp * 3 + 2] = gz;
      }
    }
#pragma unroll
    for (int c = 40; c < 48; c++) vrow[c] = 0.f;
    colbias[(b * H_ + h) * N_ + n] = -0.5f * hw * ksq + mbias;
  }
}

// ---------------------------------------------------------------------------
// Flash attention: 4 waves/block, all on the same (b,h); wave w owns i-tile.
// k'/V'/colbias tiles are staged once per block into LDS with double-buffered
// CDNA5 async global->LDS B128 copies (ASYNCcnt), overlapping the WMMA stream.
// Logits tile: 8 WMMA K-steps over augmented 32-dim vectors + column bias.
// P (C-layout) -> LDS -> A-layout; O(16x48) via 3 N-tiles x 4 K-steps WMMA.
// ---------------------------------------------------------------------------
#define TK_F 512   // 16 rows * 32 (k' tile floats)
#define TV_F 768   // 16 rows * 48 (V' tile floats)
#define TB_F 16    // colbias tile floats
#define TILE_F (TK_F + TV_F + TB_F)  // 1296

__global__ void __launch_bounds__(128) attn_kernel(
    const float* __restrict__ qa, const float* __restrict__ ka,
    const float* __restrict__ va, const float* __restrict__ colbias,
    float* __restrict__ oraw) {
  __shared__ __attribute__((aligned(16))) float sh[2 * TILE_F];  // staged tiles
  __shared__ float Pbuf[4 * 16 * 17];
  const int t = threadIdx.x;
  const int lane = t & 31;
  const int wave = t >> 5;
  const int bh = blockIdx.x;  // b*H + h
  const int b = bh / H_, h = bh % H_;
  const int i0 = (blockIdx.y * 4 + wave) * 16;
  const int lh = lane & 15, hi = lane >> 4;
  float* P = Pbuf + wave * 16 * 17;  // stride 17: conflict-free column reads

  const float* kaT = ka + bh * N_ * 32;   // 512 floats per j-tile, contiguous
  const float* vaT = va + bh * N_ * 48;   // 768 floats per j-tile, contiguous
  const float* cbT = colbias + bh * N_;   // 16 floats per j-tile

  // Cooperative async stage of tile jt into buffer buf (320 B128 + 4 B128).
  auto stage = [&](int jt, int buf) {
    float* dst = sh + buf * TILE_F;
    async_copy_b128(dst + t * 4, kaT + jt * TK_F + t * 4);            // 128xB128
    async_copy_b128(dst + TK_F + t * 4, vaT + jt * TV_F + t * 4);     // 128xB128
    if (t < 64)
      async_copy_b128(dst + TK_F + 512 + t * 4,
                      vaT + jt * TV_F + 512 + t * 4);                 // 64xB128
    if (t < 4)
      async_copy_b128(dst + TK_F + TV_F + t * 4, cbT + jt * TB_F + t * 4);
  };

  // A-fragments of q' for this row tile (resident for whole kernel)
  const float* qrow = qa + (bh * N_ + i0 + lh) * 32 + 2 * hi;
  v2f afrag[8];
#pragma unroll
  for (int s = 0; s < 8; s++) afrag[s] = *(const v2f*)(qrow + 4 * s);

  v8f o0 = {}, o1 = {}, o2 = {};
  float Mrow[8], Lrow[8];
#pragma unroll
  for (int r = 0; r < 8; r++) { Mrow[r] = -1e30f; Lrow[r] = 0.f; }

  stage(0, 0);
  wait_asynccnt0();
  __syncthreads();

  const int NT = N_ / 16;  // 24
  for (int jt = 0; jt < NT; jt++) {
    const int buf = jt & 1;
    if (jt + 1 < NT) stage(jt + 1, buf ^ 1);  // overlap DMA with compute

    const float* shK = sh + buf * TILE_F;
    const float* shV = shK + TK_F;
    const float* shC = shK + TK_F + TV_F;

    // --- logits tile: S = q' @ k'^T + colbias (k' from LDS) ---
    const float cbv = shC[lh];
    v8f sacc;
#pragma unroll
    for (int r = 0; r < 8; r++) sacc[r] = cbv;
    const float* krow = shK + lh * 32 + 2 * hi;
#pragma unroll
    for (int s = 0; s < 8; s++) {
      v2f bfrag = *(const v2f*)(krow + 4 * s);  // B[k][j] = k'[j][k]
      sacc = wmma4(afrag[s], bfrag, sacc);
    }
    // --- online softmax (row = 16 lanes of a half-wave at fixed VGPR r) ---
#pragma unroll
    for (int r = 0; r < 8; r++) {
      float v = sacc[r];
      float m = v;
      m = fmaxf(m, __shfl_xor(m, 1, 16));
      m = fmaxf(m, __shfl_xor(m, 2, 16));
      m = fmaxf(m, __shfl_xor(m, 4, 16));
      m = fmaxf(m, __shfl_xor(m, 8, 16));
      const float newM = fmaxf(Mrow[r], m);
      const float corr = __expf(Mrow[r] - newM);
      const float p = __expf(v - newM);
      float ssum = p;
      ssum += __shfl_xor(ssum, 1, 16);
      ssum += __shfl_xor(ssum, 2, 16);
      ssum += __shfl_xor(ssum, 4, 16);
      ssum += __shfl_xor(ssum, 8, 16);
      Lrow[r] = Lrow[r] * corr + ssum;
      Mrow[r] = newM;
      o0[r] *= corr; o1[r] *= corr; o2[r] *= corr;
      P[(r + 8 * hi) * 17 + lh] = p;  // C-layout -> LDS (same-wave DScnt order)
    }
    // --- O += P @ V' : P re-read in A-layout, V' B-frags from LDS ---
#pragma unroll
    for (int s = 0; s < 4; s++) {
      v2f pa;
      pa.x = P[lh * 17 + 4 * s + 2 * hi];
      pa.y = P[lh * 17 + 4 * s + 2 * hi + 1];
      const float* vr = shV + (4 * s + 2 * hi) * 48 + lh;
      v2f b0; b0.x = vr[0];  b0.y = vr[48];
      v2f b1; b1.x = vr[16]; b1.y = vr[64];
      v2f b2; b2.x = vr[32]; b2.y = vr[80];
      o0 = wmma4(pa, b0, o0);
      o1 = wmma4(pa, b1, o1);
      o2 = wmma4(pa, b2, o2);
    }
    wait_asynccnt0();  // this wave's prefetch for jt+1 done
    __syncthreads();   // all waves done reading buf / all slices landed
  }
  // --- normalize and store raw O (B,N,H,48) ---
#pragma unroll
  for (int r = 0; r < 8; r++) {
    const float inv = 1.0f / Lrow[r];
    const int i = i0 + r + 8 * hi;
    float* orow = oraw + ((b * N_ + i) * H_ + h) * 48;
    orow[lh]      = o0[r] * inv;
    orow[16 + lh] = o1[r] * inv;
    orow[32 + lh] = o2[r] * inv;
  }
}

// ---------------------------------------------------------------------------
// Finalize: inverse rigid on o_pt, norms, build cat (B,N,576)
// cat = [o(192) | o_pt.x(96) | o_pt.y(96) | o_pt.z(96) | norm(96)]
// ---------------------------------------------------------------------------
__global__ void finalize_kernel(const float* __restrict__ oraw,
                                const float* __restrict__ rot,
                                const float* __restrict__ trans,
                                float* __restrict__ cat) {
  int idx = blockIdx.x * blockDim.x + threadIdx.x;
  if (idx >= BN_) return;
  float R[9], T[3];
#pragma unroll
  for (int i = 0; i < 9; i++) R[i] = rot[idx * 9 + i];
#pragma unroll
  for (int i = 0; i < 3; i++) T[i] = trans[idx * 3 + i];
  float* crow = cat + idx * 576;
  for (int h = 0; h < H_; h++) {
    const float* orow = oraw + (idx * H_ + h) * 48;
#pragma unroll
    for (int c = 0; c < 16; c++) crow[h * 16 + c] = orow[c];
#pragma unroll
    for (int p = 0; p < PV_; p++) {
      const float x = orow[16 + p * 3 + 0] - T[0];
      const float y = orow[16 + p * 3 + 1] - T[1];
      const float z = orow[16 + p * 3 + 2] - T[2];
      const float ox = R[0] * x + R[3] * y + R[6] * z;  // R^T
      const float oy = R[1] * x + R[4] * y + R[7] * z;
      const float oz = R[2] * x + R[5] * y + R[8] * z;
      crow[192 + h * PV_ + p] = ox;
      crow[288 + h * PV_ + p] = oy;
      crow[384 + h * PV_ + p] = oz;
      crow[480 + h * PV_ + p] = sqrtf(ox * ox + oy * oy + oz * oz + 1e-8f);
    }
  }
}

// ---------------------------------------------------------------------------
extern "C" void kernel_launch(void* const* d_in, const int* in_sizes, int n_in,
                              void* d_out, int out_size, void* d_ws,
                              size_t ws_size, hipStream_t stream) {
  (void)in_sizes; (void)n_in; (void)out_size; (void)ws_size;
  const float* s     = (const float*)d_in[0];
  // d_in[1] = z : dead in reference (del z) -> never read
  const float* rot   = (const float*)d_in[2];
  const float* trans = (const float*)d_in[3];
  const float* mask  = (const float*)d_in[4];
  const float* w_q   = (const float*)d_in[5];
  const float* b_q   = (const float*)d_in[6];
  const float* w_kv  = (const float*)d_in[7];
  const float* b_kv  = (const float*)d_in[8];
  // d_in[9..12] = w_gq/b_gq/w_gk/b_gk : dead in reference
  const float* w_qp  = (const float*)d_in[13];
  const float* b_qp  = (const float*)d_in[14];
  const float* w_kvp = (const float*)d_in[15];
  const float* b_kvp = (const float*)d_in[16];
  const float* hwts  = (const float*)d_in[17];
  const float* w_out = (const float*)d_in[18];
  const float* b_out = (const float*)d_in[19];

  float* ws = (float*)d_ws;
  // Region A: q (294912) + kv (589824)  -> later reused for cat (884736)
  float* q       = ws;
  float* kv      = ws + 294912;
  // Region B: qp_raw (221184) + kvp_raw (663552) -> later reused for oraw
  float* qp_raw  = ws + 884736;
  float* kvp_raw = ws + 1105920;
  float* qa      = ws + 1769472;  // B*H*N*32 = 589824
  float* ka      = ws + 2359296;  // 589824
  float* va      = ws + 2949120;  // B*H*N*48 = 884736
  float* cb      = ws + 3833856;  // B*H*N    = 18432   (total 3852288 floats)
  float* oraw    = qp_raw;        // alias region B (dead after prep)
  float* cat     = ws;            // alias region A (dead after attention)

  dim3 blk(128);
  // Projections: s @ W + b   (M=1536, K=384)
  gemm16_bias<<<dim3(96, 3), blk, 0, stream>>>(s, w_q,   b_q,   q,       384, 192);
  gemm16_bias<<<dim3(96, 6), blk, 0, stream>>>(s, w_kv,  b_kv,  kv,      384, 384);
  gemm16_bias<<<dim3(96, 3), blk, 0, stream>>>(s, w_qp,  b_qp,  qp_raw,  384, 144);
  gemm16_bias<<<dim3(96, 7), blk, 0, stream>>>(s, w_kvp, b_kvp, kvp_raw, 384, 432);
  // Rigid apply + augmented vectors
  prep_kernel<<<6, 256, 0, stream>>>(q, kv, qp_raw, kvp_raw, rot, trans, mask,
                                     hwts, qa, ka, va, cb);
  // Fused attention (B*H = 48 heads, 24 row-tiles)
  attn_kernel<<<dim3(48, 6), blk, 0, stream>>>(qa, ka, va, cb, oraw);
  // Inverse rigid + concat
  finalize_kernel<<<6, 256, 0, stream>>>(oraw, rot, trans, cat);
  // Output projection: cat @ w_out + b_out  (K=576, Nout=384)
  gemm16_bias<<<dim3(96, 6), blk, 0, stream>>>(cat, w_out, b_out, (float*)d_out,
                                               576, 384);
}